// MFVISemanticDependency_36653250904395
// MI455X (gfx1250) — compile-verified
//
#include <hip/hip_runtime.h>
#include <math.h>

// MI455X / gfx1250: wave32, WMMA. f32 path uses V_WMMA_F32_16X16X4_F32.
typedef __attribute__((ext_vector_type(2))) float v2f;
typedef __attribute__((ext_vector_type(8))) float v8f;

#define SDIM 160
#define BATCH 4
#define BSS (BATCH * SDIM * SDIM)   // elements of one [B,S,S] f32 plane

// ---------------------------------------------------------------------------
// init: Q0 = sigmoid(0) = 0.5 everywhere;  W[b,h,t] = mask[b,t,h] * (t != h)
// ---------------------------------------------------------------------------
__global__ void mfvi_init_kernel(const unsigned char* __restrict__ mask,
                                 float* __restrict__ Q0,
                                 float* __restrict__ W) {
  int idx = blockIdx.x * blockDim.x + threadIdx.x;
  if (idx >= BSS) return;
  int t = idx % SDIM;
  int h = (idx / SDIM) % SDIM;
  int b = idx / (SDIM * SDIM);
  Q0[idx] = 0.5f;
  W[idx] = (t != h && mask[(b * SDIM + t) * SDIM + h]) ? 1.0f : 0.0f;
}

// ---------------------------------------------------------------------------
// One mean-field iteration. One workgroup per (b,m); 10 waves, one 16-wide
// h-tile per wave; t-contraction (K=160) done as 40 steps of K=4 f32 WMMA.
//   A (16x4) = q1 vector chunk broadcast over M  -> all C rows identical
//   B (4x16) = tensor tile [t,h] (one global_load_b64 per lane)
//   C row 0  = f[h0 + (lane&15)] replicated in every lane
// ---------------------------------------------------------------------------
__global__ __launch_bounds__(320)
void mfvi_iter_kernel(const float* __restrict__ s_edge,
                      const float* __restrict__ s_sib,
                      const float* __restrict__ s_cop,
                      const float* __restrict__ s_grd,
                      const unsigned char* __restrict__ mask,
                      const float* __restrict__ Qin,   // [b,h,t] = q1[h,t,b]
                      const float* __restrict__ W,     // [b,h,t]
                      float* __restrict__ Qout,
                      float* __restrict__ out,         // [b,m,h,2]
                      int last) {
  __shared__ float v2s[SDIM];   // q1[t,m,b] * (t!=m)   (column m of Q)
  __shared__ float v3s[SDIM];   // q1[m,t,b] * (t!=m)   (row m of Q)

  const int bm = blockIdx.x;            // b*S + m
  const int b  = bm / SDIM;
  const int m  = bm % SDIM;
  const int tid = threadIdx.x;

  const float* Qb = Qin + (size_t)b * SDIM * SDIM;
  const float* Wb = W   + (size_t)b * SDIM * SDIM;

  for (int t = tid; t < SDIM; t += 320) {
    float z = (t == m) ? 0.0f : 1.0f;
    v3s[t] = Qb[m * SDIM + t] * z;
    v2s[t] = Qb[t * SDIM + m] * z;
  }
  __syncthreads();

  const int lane = tid & 31;
  const int wid  = tid >> 5;            // 0..9
  const int h0   = wid * 16;
  const int n    = lane & 15;           // column (h offset) this lane owns
  const int kb   = (lane >> 4) * 2;     // 0 for lanes 0-15, 2 for lanes 16-31

  const size_t tilebase = (size_t)bm * SDIM * SDIM;   // [b,m,:,:] block
  const int rowoff = (h0 + n) * SDIM + kb;
  const float* psib = s_sib + tilebase + rowoff;
  const float* pcop = s_cop + tilebase + rowoff;
  const float* pgrd = s_grd + tilebase + rowoff;
  const float* pq   = Qb + rowoff;
  const float* pw   = Wb + rowoff;

  v8f acc = {0.f, 0.f, 0.f, 0.f, 0.f, 0.f, 0.f, 0.f};
  v2f ones; ones.x = 1.0f; ones.y = 1.0f;

  for (int t0 = 0; t0 < SDIM; t0 += 4) {
    // B tiles: lane (n, kb-half) -> elements (t0+kb, t0+kb+1), 8B aligned
    v2f bs = *(const v2f*)(psib + t0);
    v2f bc = *(const v2f*)(pcop + t0);
    v2f bg = *(const v2f*)(pgrd + t0);
    v2f qt = *(const v2f*)(pq + t0);    // q1[h, t] tile (term 1)
    v2f wt = *(const v2f*)(pw + t0);    // (t!=h)*mask[b,t,h] tile
    // A vectors: broadcast over M (rows)
    v2f a2; a2.x = v2s[t0 + kb]; a2.y = v2s[t0 + kb + 1];
    v2f a3; a3.x = v3s[t0 + kb]; a3.y = v3s[t0 + kb + 1];

    bc.x *= wt.x; bc.y *= wt.y;
    bg.x *= wt.x; bg.y *= wt.y;
    bs.x *= wt.x * qt.x; bs.y *= wt.y * qt.y;

    acc = __builtin_amdgcn_wmma_f32_16x16x4_f32(false, a2,   false, bc,
                                                (short)0, acc, false, false);
    acc = __builtin_amdgcn_wmma_f32_16x16x4_f32(false, a3,   false, bg,
                                                (short)0, acc, false, false);
    acc = __builtin_amdgcn_wmma_f32_16x16x4_f32(false, ones, false, bs,
                                                (short)0, acc, false, false);
  }

  // Every lane: acc[0] = C[M, N=lane&15] = f[h0 + (lane&15)] (rows identical)
  int h = h0 + n;
  float f = acc[0];
  // Remove t==m contribution that leaked into term 1 (v2/v3 already zeroed)
  f -= Qb[h * SDIM + m] * s_sib[tilebase + h * SDIM + m] * Wb[h * SDIM + m];
  // mask[b,m,h] multiplies all second-order terms
  f *= (float)mask[(b * SDIM + m) * SDIM + h];

  float x  = s_edge[(b * SDIM + m) * SDIM + h] + f;
  float sp = 1.0f / (1.0f + __expf(-x));   // sigmoid(x)

  if (lane < 16) {                          // lanes 16-31 hold duplicates
    if (last) {
      size_t o = ((size_t)(b * SDIM + m) * SDIM + h) * 2;
      out[o]     = 1.0f / (1.0f + __expf(x));   // sigmoid(-x)
      out[o + 1] = sp;
    } else {
      Qout[(size_t)b * SDIM * SDIM + h * SDIM + m] = sp;
    }
  }
}

// ---------------------------------------------------------------------------
extern "C" void kernel_launch(void* const* d_in, const int* in_sizes, int n_in,
                              void* d_out, int out_size, void* d_ws, size_t ws_size,
                              hipStream_t stream) {
  const float* s_edge = (const float*)d_in[0];
  const float* s_sib  = (const float*)d_in[1];
  const float* s_cop  = (const float*)d_in[2];
  const float* s_grd  = (const float*)d_in[3];
  const unsigned char* mask = (const unsigned char*)d_in[4];
  float* out = (float*)d_out;

  float* Q0 = (float*)d_ws;     // [B,S,S]
  float* Q1 = Q0 + BSS;         // [B,S,S]
  float* W  = Q1 + BSS;         // [B,S,S]

  mfvi_init_kernel<<<(BSS + 255) / 256, 256, 0, stream>>>(mask, Q0, W);

  // MAX_ITER = 3, Q ping-pong
  mfvi_iter_kernel<<<BATCH * SDIM, 320, 0, stream>>>(
      s_edge, s_sib, s_cop, s_grd, mask, Q0, W, Q1, out, 0);
  mfvi_iter_kernel<<<BATCH * SDIM, 320, 0, stream>>>(
      s_edge, s_sib, s_cop, s_grd, mask, Q1, W, Q0, out, 0);
  mfvi_iter_kernel<<<BATCH * SDIM, 320, 0, stream>>>(
      s_edge, s_sib, s_cop, s_grd, mask, Q0, W, Q1, out, 1);
}